// P_42803644072813
// MI455X (gfx1250) — compile-verified
//
#include <hip/hip_runtime.h>
#include <cstdint>

// ---------------------------------------------------------------------------
// Hyperelastic P = dW/dF for batched 3x3 F (closed-form gradient).
// Bandwidth-bound: 144 MB total @ 23.3 TB/s ~= 6.2 us floor; compute is
// ~0.3 GFLOP, far below VALU capability, so the win is the data path.
// MI455X-idiomatic movement: one Tensor-Data-Mover DMA per 9216 B tile
// (tensor_load_to_lds / tensor_store_from_lds, TENSORcnt). This toolchain
// exposes the 6-arg builtin form (clang-23 / therock-10.0 per compile probe).
// ---------------------------------------------------------------------------

constexpr int TPB  = 256;        // 8 waves (wave32), 1 sample per lane
constexpr int FPB  = TPB * 9;    // floats per block tile = 2304 (9216 B)
constexpr int V4PB = FPB / 4;    // float4 chunks per tile = 576

#if defined(__HIP_DEVICE_COMPILE__) && defined(__gfx1250__)
  #define IS_GFX1250_DEV 1
  #if __has_builtin(__builtin_amdgcn_tensor_load_to_lds) && \
      __has_builtin(__builtin_amdgcn_tensor_store_from_lds)
    #define HAVE_TDM 1
  #else
    #define HAVE_TDM 0
  #endif
#else
  #define IS_GFX1250_DEV 0
  #define HAVE_TDM 0
#endif

#if HAVE_TDM
typedef unsigned int u32x4 __attribute__((ext_vector_type(4)));
typedef int          i32x8 __attribute__((ext_vector_type(8)));
typedef int          i32x4 __attribute__((ext_vector_type(4)));

// Build a 2D D# describing one contiguous tile of 4-byte elements:
//   tensor_dim0 = nValid (clips the ragged tail: OOB reads->0, OOB writes drop)
//   tile_dim0   = FPB elements, tensor_dim1 = tile_dim1 = 1
__device__ __forceinline__ void tdm_tile(bool isStore, uint32_t ldsByteAddr,
                                         const void* gptr, uint32_t nValid) {
  const uint64_t ga = (uint64_t)(uintptr_t)gptr;
  u32x4 g0;
  g0.x = 1u;                                            // count=1, user D#
  g0.y = ldsByteAddr;                                   // lds_addr (bytes)
  g0.z = (uint32_t)(ga & 0xFFFFFFFFu);                  // global_addr[31:0]
  g0.w = (uint32_t)((ga >> 32) & 0x01FFFFFFu)           // global_addr[56:32]
       | (2u << 30);                                    // type = 2 ("image")
  i32x8 g1;
  g1[0] = (int)(2u << 16);                              // data_size = 4 B
  g1[1] = (int)((nValid & 0xFFFFu) << 16);              // tensor_dim0[15:0]
  g1[2] = (int)((nValid >> 16) | (1u << 16));           // dim0[31:16], tensor_dim1=1
  g1[3] = (int)((uint32_t)FPB << 16);                   // dim1 hi=0, tile_dim0=2304
  g1[4] = 1;                                            // tile_dim1=1, tile_dim2=0
  g1[5] = (int)nValid;                                  // tensor_dim0_stride[31:0]
  g1[6] = 0;                                            // stride0 hi, stride1 lo
  g1[7] = 0;                                            // stride1 hi
  const i32x4 z4 = {0, 0, 0, 0};                        // groups 2/3: <=2D tensor
  const i32x8 z8 = {0, 0, 0, 0, 0, 0, 0, 0};           // extended words unused
  if (isStore) __builtin_amdgcn_tensor_store_from_lds(g0, g1, z4, z4, z8, 0);
  else         __builtin_amdgcn_tensor_load_to_lds(g0, g1, z4, z4, z8, 0);
}
#endif // HAVE_TDM

__device__ __forceinline__ void wait_tdm() {
#if IS_GFX1250_DEV
  #if __has_builtin(__builtin_amdgcn_s_wait_tensorcnt)
    __builtin_amdgcn_s_wait_tensorcnt(0);
  #else
    asm volatile("s_wait_tensorcnt 0x0" ::: "memory");
  #endif
#endif
}

// --- fallback path (probe-verified in round 1) -----------------------------
__device__ __forceinline__ void async_load_f4_to_lds(float4* lds_dst,
                                                     const float4* gsrc) {
#if IS_GFX1250_DEV
  uint32_t lds_addr = (uint32_t)(uintptr_t)lds_dst;
  asm volatile("global_load_async_to_lds_b128 %0, %1, off"
               :
               : "v"(lds_addr), "v"(gsrc)
               : "memory");
#else
  *lds_dst = *gsrc;
#endif
}

__device__ __forceinline__ void wait_async_lds() {
#if IS_GFX1250_DEV
  asm volatile("s_wait_asynccnt 0" ::: "memory");
#endif
}

__global__ __launch_bounds__(TPB) void hyper_grad_kernel(
    const float* __restrict__ F, float* __restrict__ P,
    int N, int totalFloats) {
  __shared__ float4 sA4[V4PB];   // staged input tile  (9216 B)
  __shared__ float4 sB4[V4PB];   // staged output tile (9216 B)
  float* sA = (float*)sA4;
  float* sB = (float*)sB4;

  const int tid            = threadIdx.x;
  const int blockFloatBase = blockIdx.x * FPB;   // < 18M, fits int
  const int remain         = totalFloats - blockFloatBase;
  const uint32_t nValid    = (uint32_t)(remain < FPB ? remain : FPB);

  // TDM ignores EXEC, so the one-wave issue MUST be a scalar branch (an
  // EXEC-masked issue from all 8 waves would DMA the tile 8x). readfirstlane
  // forces the condition into an SGPR -> s_cmp/s_cbranch.
  const int waveId = __builtin_amdgcn_readfirstlane((int)(threadIdx.x >> 5));

  // ---- stage input tile into LDS ----
#if HAVE_TDM
  if (waveId == 0) {
    tdm_tile(false, (uint32_t)(uintptr_t)sA4, (const void*)(F + blockFloatBase),
             nValid);
    wait_tdm();                    // tile resident before the barrier
  }
#else
  {
    const int base4 = blockFloatBase >> 2;
    for (int i = tid; i < V4PB; i += TPB) {
      const int g4 = base4 + i;
      const int f0 = g4 << 2;
      if (f0 + 4 <= totalFloats) {
        async_load_f4_to_lds(&sA4[i], ((const float4*)F) + g4);
      } else if (f0 < totalFloats) {
        for (int k = 0; f0 + k < totalFloats; ++k) sA[i * 4 + k] = F[f0 + k];
      }
    }
    wait_async_lds();
  }
#endif
  // Speculative prefetch of the next tile toward L2 (global_prefetch_b8).
  {
    const long nb = (long)(blockFloatBase + FPB) * 4 + (long)tid * 64;
    if (tid < (FPB * 4) / 64 && nb < (long)totalFloats * 4)
      __builtin_prefetch(((const char*)F) + nb, 0, 1);
  }
  __syncthreads();

  // ---- per-sample closed-form gradient (one sample per lane) ----
  const int s = blockIdx.x * TPB + tid;
  if (s < N) {
    // 9-word lane stride is coprime with 64 LDS banks -> conflict-free.
    const float* f = sA + tid * 9;
    const float F00 = f[0], F01 = f[1], F02 = f[2];
    const float F10 = f[3], F11 = f[4], F12 = f[5];
    const float F20 = f[6], F21 = f[7], F22 = f[8];

    // C = F^T F (symmetric)
    const float C00 = F00*F00 + F10*F10 + F20*F20;
    const float C01 = F00*F01 + F10*F11 + F20*F21;
    const float C02 = F00*F02 + F10*F12 + F20*F22;
    const float C11 = F01*F01 + F11*F11 + F21*F21;
    const float C12 = F01*F02 + F11*F12 + F21*F22;
    const float C22 = F02*F02 + F12*F12 + F22*F22;

    // cof(F) = dJ/dF, J = det F
    const float cof00 = F11*F22 - F21*F12;
    const float cof10 = F21*F02 - F01*F22;
    const float cof20 = F01*F12 - F11*F02;
    const float cof01 = F12*F20 - F22*F10;
    const float cof11 = F22*F00 - F02*F20;
    const float cof21 = F02*F10 - F12*F00;
    const float cof02 = F10*F21 - F20*F11;
    const float cof12 = F20*F01 - F00*F21;
    const float cof22 = F00*F11 - F10*F01;
    const float J = F00*cof00 + F10*cof10 + F20*cof20;

    const float rJ  = 1.0f / J;
    const float I3  = J * J;
    const float rI3 = rJ * rJ;
    const float sJ  = 20.0f * J - 56.0f * rJ;   // d(10J^2 - 56 log J)/dJ

    // adj(C) (symmetric), Cinv = adj(C)/I3
    const float A00 = C11*C22 - C12*C12;
    const float A01 = C02*C12 - C01*C22;
    const float A02 = C01*C12 - C02*C11;
    const float A11 = C00*C22 - C02*C02;
    const float A12 = C01*C02 - C00*C12;
    const float A22 = C00*C11 - C01*C01;
    const float Ci00 = A00*rI3, Ci01 = A01*rI3, Ci02 = A02*rI3;
    const float Ci11 = A11*rI3, Ci12 = A12*rI3, Ci22 = A22*rI3;

    const float g0 = 4.0f, g1 = 0.5f, g2 = 0.5f;     // G = diag(4,.5,.5)
    const float I4 = g0*C00  + g1*C11  + g2*C22;     // tr(C G)
    const float t  = g0*Ci00 + g1*Ci11 + g2*Ci22;    // tr(Cinv G)
    const float I5 = I3 * t;                         // tr(Cof(C) G)

    // K = Cinv G Cinv (symmetric)
    const float K00 = g0*Ci00*Ci00 + g1*Ci01*Ci01 + g2*Ci02*Ci02;
    const float K01 = g0*Ci00*Ci01 + g1*Ci01*Ci11 + g2*Ci02*Ci12;
    const float K02 = g0*Ci00*Ci02 + g1*Ci01*Ci12 + g2*Ci02*Ci22;
    const float K11 = g0*Ci01*Ci01 + g1*Ci11*Ci11 + g2*Ci12*Ci12;
    const float K12 = g0*Ci01*Ci02 + g1*Ci11*Ci12 + g2*Ci12*Ci22;
    const float K22 = g0*Ci02*Ci02 + g1*Ci12*Ci12 + g2*Ci22*Ci22;

    // S5 = dI5/dC = I5*Cinv - I3*K
    const float S500 = I5*Ci00 - I3*K00;
    const float S501 = I5*Ci01 - I3*K01;
    const float S502 = I5*Ci02 - I3*K02;
    const float S511 = I5*Ci11 - I3*K11;
    const float S512 = I5*Ci12 - I3*K12;
    const float S522 = I5*Ci22 - I3*K22;

    // M = 16*I + I4*G + I5*S5 ;  P = F*M + sJ*cof(F)
    const float M00 = 16.0f + g0*I4 + I5*S500;
    const float M01 =                 I5*S501;
    const float M02 =                 I5*S502;
    const float M11 = 16.0f + g1*I4 + I5*S511;
    const float M12 =                 I5*S512;
    const float M22 = 16.0f + g2*I4 + I5*S522;

    float* o = sB + tid * 9;
    o[0] = F00*M00 + F01*M01 + F02*M02 + sJ*cof00;
    o[1] = F00*M01 + F01*M11 + F02*M12 + sJ*cof01;
    o[2] = F00*M02 + F01*M12 + F02*M22 + sJ*cof02;
    o[3] = F10*M00 + F11*M01 + F12*M02 + sJ*cof10;
    o[4] = F10*M01 + F11*M11 + F12*M12 + sJ*cof11;
    o[5] = F10*M02 + F11*M12 + F12*M22 + sJ*cof12;
    o[6] = F20*M00 + F21*M01 + F22*M02 + sJ*cof20;
    o[7] = F20*M01 + F21*M11 + F22*M12 + sJ*cof21;
    o[8] = F20*M02 + F21*M12 + F22*M22 + sJ*cof22;
  }
  __syncthreads();

  // ---- drain output tile ----
#if HAVE_TDM
  if (waveId == 0) {
    tdm_tile(true, (uint32_t)(uintptr_t)sB4, (void*)(P + blockFloatBase),
             nValid);
    wait_tdm();   // keep workgroup (and its LDS) alive until the DMA drains
  }
#else
  {
    const int base4 = blockFloatBase >> 2;
    for (int i = tid; i < V4PB; i += TPB) {
      const int g4 = base4 + i;
      const int f0 = g4 << 2;
      if (f0 + 4 <= totalFloats) {
        ((float4*)P)[g4] = sB4[i];
      } else if (f0 < totalFloats) {
        for (int k = 0; f0 + k < totalFloats; ++k) P[f0 + k] = sB[i * 4 + k];
      }
    }
  }
#endif
}

extern "C" void kernel_launch(void* const* d_in, const int* in_sizes, int n_in,
                              void* d_out, int out_size, void* d_ws, size_t ws_size,
                              hipStream_t stream) {
  (void)n_in; (void)out_size; (void)d_ws; (void)ws_size;
  const float* F = (const float*)d_in[0];
  float* P = (float*)d_out;
  const int totalFloats = in_sizes[0];     // N*9 = 18,000,000
  const int N = totalFloats / 9;
  const int blocks = (N + TPB - 1) / TPB;  // one sample per thread
  hipLaunchKernelGGL(hyper_grad_kernel, dim3(blocks), dim3(TPB), 0, stream,
                     F, P, N, totalFloats);
}